// AccessibilitySVIGNN_42777874268501
// MI455X (gfx1250) — compile-verified
//
#include <hip/hip_runtime.h>
#include <hip/hip_bf16.h>

typedef __attribute__((ext_vector_type(2))) float v2f;
typedef __attribute__((ext_vector_type(8))) float v8f;

#define LN_EPS 1e-5f

// ---------------------------------------------------------------------------
// Utility kernels
// ---------------------------------------------------------------------------
__global__ void zero_kernel(float* __restrict__ p, long n) {
  long i = (long)blockIdx.x * blockDim.x + threadIdx.x;
  if (i < n) p[i] = 0.f;
}

__global__ void init_deg_kernel(float* __restrict__ deg, int n) {
  int i = blockIdx.x * blockDim.x + threadIdx.x;
  if (i < n) deg[i] = 1.0f;  // self-loop contributes 1
}

__global__ void count_deg_kernel(const int* __restrict__ dst, float* __restrict__ deg, int e) {
  int i = blockIdx.x * blockDim.x + threadIdx.x;
  if (i < e) atomicAdd(&deg[dst[i]], 1.0f);
}

__device__ inline void atomicMaxF(float* addr, float val) {
  unsigned int* ua = reinterpret_cast<unsigned int*>(addr);
  unsigned int old = __float_as_uint(*addr);
  while (__uint_as_float(old) < val) {
    unsigned int prev = atomicCAS(ua, old, __float_as_uint(val));
    if (prev == old) break;
    old = prev;
  }
}

// ---------------------------------------------------------------------------
// Context gate + input LayerNorm  (per-node, weights staged in LDS)
// ---------------------------------------------------------------------------
__global__ __launch_bounds__(256) void ctx_gate_ln_kernel(
    const float* __restrict__ acc, const float* __restrict__ ctx,
    const float* __restrict__ cg_w1, const float* __restrict__ cg_b1,
    const float* __restrict__ cg_w2, const float* __restrict__ cg_b2,
    const float* __restrict__ cg_aw, const float* __restrict__ cg_ab,
    const float* __restrict__ base_imp,
    const float* __restrict__ ln_g, const float* __restrict__ ln_b,
    float* __restrict__ xln, int n) {
  __shared__ float sW1[5 * 32], sW2[32 * 32], sAW[32 * 32];
  __shared__ float sB1[32], sB2[32], sAB[32], sBI[32], sLG[32], sLB[32];
  int tid = threadIdx.x;
  for (int i = tid; i < 5 * 32; i += 256) sW1[i] = cg_w1[i];
  for (int i = tid; i < 32 * 32; i += 256) { sW2[i] = cg_w2[i]; sAW[i] = cg_aw[i]; }
  if (tid < 32) {
    sB1[tid] = cg_b1[tid]; sB2[tid] = cg_b2[tid]; sAB[tid] = cg_ab[tid];
    sBI[tid] = base_imp[tid]; sLG[tid] = ln_g[tid]; sLB[tid] = ln_b[tid];
  }
  __syncthreads();
  int i = blockIdx.x * 256 + tid;
  if (i >= n) return;

  float c[5];
#pragma unroll
  for (int k = 0; k < 5; k++) c[k] = ctx[i * 5 + k];

  float h1[32];
#pragma unroll
  for (int j = 0; j < 32; j++) {
    float s = sB1[j];
#pragma unroll
    for (int k = 0; k < 5; k++) s += c[k] * sW1[k * 32 + j];
    h1[j] = fmaxf(s, 0.f);
  }
  float h2[32];
#pragma unroll
  for (int j = 0; j < 32; j++) {
    float s = sB2[j];
#pragma unroll
    for (int k = 0; k < 32; k++) s += h1[k] * sW2[k * 32 + j];
    h2[j] = s;
  }
  float lg[32];
  float mx = -1e30f;
#pragma unroll
  for (int j = 0; j < 32; j++) {
    float s = sAB[j];
#pragma unroll
    for (int k = 0; k < 32; k++) s += h2[k] * sAW[k * 32 + j];
    lg[j] = s;
    mx = fmaxf(mx, s);
  }
  float ssum = 0.f;
#pragma unroll
  for (int j = 0; j < 32; j++) { lg[j] = __expf(lg[j] - mx); ssum += lg[j]; }
  float inv = 1.f / ssum;

  float x[32];
  float mu = 0.f;
#pragma unroll
  for (int j = 0; j < 32; j++) {
    x[j] = acc[i * 32 + j] * (lg[j] * inv) * sBI[j];
    mu += x[j];
  }
  mu *= (1.f / 32.f);
  float var = 0.f;
#pragma unroll
  for (int j = 0; j < 32; j++) { float d = x[j] - mu; var += d * d; }
  var *= (1.f / 32.f);
  float rstd = rsqrtf(var + LN_EPS);
#pragma unroll
  for (int j = 0; j < 32; j++)
    xln[i * 32 + j] = (x[j] - mu) * rstd * sLG[j] + sLB[j];
}

// ---------------------------------------------------------------------------
// WMMA f32 GEMM: out[n,NCOL] = epilogue(prelogue(A)[n,K] @ W[K,NCOL])
//  - One wave computes a 16x16 tile with V_WMMA_F32_16X16X4_F32, K step 4.
//  - W staged in LDS; fragment layouts per ISA 7.12.2 (f32).
//  - PRE_BN: fuse y = relu((a + cb - m) * g*rsqrt(v+eps) + b) into the
//    A-fragment load (per-input-channel scale/shift staged in LDS).
//  - Epilogue: optional bias + relu; branch-free full-tile fast path.
// ---------------------------------------------------------------------------
template <int K, int NCOL, bool RELU, bool BIAS, bool PRE_BN>
__global__ __launch_bounds__(256) void gemm_wmma_kernel(
    const float* __restrict__ A, const float* __restrict__ W,
    const float* __restrict__ bias, float* __restrict__ out, int n,
    const float* __restrict__ cb, const float* __restrict__ bng,
    const float* __restrict__ bnb, const float* __restrict__ bnm,
    const float* __restrict__ bnv) {
  __shared__ float sW[K * NCOL];
  __shared__ float sS[K], sT[K];
  int tid = threadIdx.x;
  for (int i = tid; i < K * NCOL; i += 256) sW[i] = W[i];
  if (PRE_BN) {
    if (tid < K) {
      float s = bng[tid] * rsqrtf(bnv[tid] + LN_EPS);
      sS[tid] = s;
      sT[tid] = (cb[tid] - bnm[tid]) * s + bnb[tid];
    }
  }
  __syncthreads();

  const int colTiles = NCOL / 16;
  int rowTiles = (n + 15) >> 4;
  int wave = (blockIdx.x * 256 + tid) >> 5;
  int lane = tid & 31;
  if (wave >= rowTiles * colTiles) return;  // wave-uniform: EXEC stays all-1s

  int rowTile = wave / colTiles;
  int colTile = wave - rowTile * colTiles;
  int row0 = rowTile << 4;
  int col0 = colTile << 4;

  int nn = lane & 15;   // M for A-frag, N for B/C-frag
  int kh = lane >> 4;   // K half-selector

  int r = row0 + nn;
  int rc = r < n ? r : n - 1;  // clamped load keeps EXEC full
  const float* __restrict__ Arow = A + (size_t)rc * K;

  v8f cacc = {};
#pragma unroll
  for (int k0 = 0; k0 < K; k0 += 4) {
    int ka = k0 + 2 * kh;
    v2f a, b;
    float ax = Arow[ka];
    float ay = Arow[ka + 1];
    if (PRE_BN) {
      ax = fmaxf(ax * sS[ka] + sT[ka], 0.f);
      ay = fmaxf(ay * sS[ka + 1] + sT[ka + 1], 0.f);
    }
    a.x = ax;
    a.y = ay;
    b.x = sW[ka * NCOL + col0 + nn];
    b.y = sW[(ka + 1) * NCOL + col0 + nn];
    cacc = __builtin_amdgcn_wmma_f32_16x16x4_f32(
        false, a, false, b, (short)0, cacc, false, false);
  }

  float bv = 0.f;
  if (BIAS) bv = bias[col0 + nn];
  float* __restrict__ outc = out + col0 + nn;
  if (row0 + 16 <= n) {  // full tile: wave-uniform, no per-row branches
#pragma unroll
    for (int i = 0; i < 8; i++) {
      int row = row0 + i + 8 * kh;
      float v = cacc[i] + bv;
      if (RELU) v = fmaxf(v, 0.f);
      outc[(size_t)row * NCOL] = v;
    }
  } else {
#pragma unroll
    for (int i = 0; i < 8; i++) {
      int row = row0 + i + 8 * kh;
      if (row < n) {
        float v = cacc[i] + bv;
        if (RELU) v = fmaxf(v, 0.f);
        outc[(size_t)row * NCOL] = v;
      }
    }
  }
}

// ---------------------------------------------------------------------------
// GCN edge scatter: out[dst] += h[src] * D^-1/2[src] * D^-1/2[dst]
// logC = 6 (H=64) or 5 (H=32). Edges e>=E are the N self-loops.
// ---------------------------------------------------------------------------
__global__ void gcn_scatter_kernel(const float* __restrict__ h,
                                   const int* __restrict__ src,
                                   const int* __restrict__ dst,
                                   const float* __restrict__ deg,
                                   float* __restrict__ out, int E, int n, int logC) {
  long t = (long)blockIdx.x * blockDim.x + threadIdx.x;
  int C = 1 << logC;
  long total = ((long)E + n) << logC;
  if (t >= total) return;
  int e = (int)(t >> logC);
  int c = (int)(t & (C - 1));
  int s, d;
  if (e < E) { s = src[e]; d = dst[e]; } else { s = d = e - E; }
  float coef = rsqrtf(deg[s] * deg[d]);
  atomicAdd(&out[((long)d << logC) + c], h[((long)s << logC) + c] * coef);
}

// ---------------------------------------------------------------------------
// GAT pieces
// ---------------------------------------------------------------------------
__global__ void gat_alpha_kernel(const float* __restrict__ h,
                                 const float* __restrict__ asrc,
                                 const float* __restrict__ adst,
                                 float* __restrict__ as_, float* __restrict__ ad_, int n) {
  int i = blockIdx.x * blockDim.x + threadIdx.x;
  if (i >= n) return;
#pragma unroll
  for (int hd = 0; hd < 2; hd++) {
    float s = 0.f, d = 0.f;
#pragma unroll
    for (int c = 0; c < 32; c++) {
      float hv = h[i * 64 + hd * 32 + c];
      s += hv * asrc[hd * 32 + c];
      d += hv * adst[hd * 32 + c];
    }
    as_[i * 2 + hd] = s;
    ad_[i * 2 + hd] = d;
  }
}

__global__ void gat_init_kernel(float* __restrict__ mx, float* __restrict__ sm, int n2) {
  int i = blockIdx.x * blockDim.x + threadIdx.x;
  if (i < n2) { mx[i] = -1e30f; sm[i] = 0.f; }
}

__device__ inline float gat_logit(const float* as_, const float* ad_, int s, int d, int hd) {
  float e = as_[s * 2 + hd] + ad_[d * 2 + hd];
  return e >= 0.f ? e : 0.2f * e;  // leaky_relu(., 0.2)
}

__global__ void gat_max_kernel(const int* __restrict__ src, const int* __restrict__ dst,
                               const float* __restrict__ as_, const float* __restrict__ ad_,
                               float* __restrict__ mx, int E, int n) {
  long t = (long)blockIdx.x * blockDim.x + threadIdx.x;
  long total = ((long)E + n) * 2;
  if (t >= total) return;
  int e = (int)(t >> 1), hd = (int)(t & 1);
  int s, d;
  if (e < E) { s = src[e]; d = dst[e]; } else { s = d = e - E; }
  atomicMaxF(&mx[d * 2 + hd], gat_logit(as_, ad_, s, d, hd));
}

__global__ void gat_sum_kernel(const int* __restrict__ src, const int* __restrict__ dst,
                               const float* __restrict__ as_, const float* __restrict__ ad_,
                               const float* __restrict__ mx, float* __restrict__ sm,
                               int E, int n) {
  long t = (long)blockIdx.x * blockDim.x + threadIdx.x;
  long total = ((long)E + n) * 2;
  if (t >= total) return;
  int e = (int)(t >> 1), hd = (int)(t & 1);
  int s, d;
  if (e < E) { s = src[e]; d = dst[e]; } else { s = d = e - E; }
  float ev = gat_logit(as_, ad_, s, d, hd);
  atomicAdd(&sm[d * 2 + hd], __expf(ev - mx[d * 2 + hd]));
}

__global__ void gat_scatter_kernel(const float* __restrict__ h,
                                   const int* __restrict__ src, const int* __restrict__ dst,
                                   const float* __restrict__ as_, const float* __restrict__ ad_,
                                   const float* __restrict__ mx, const float* __restrict__ sm,
                                   float* __restrict__ out, int E, int n) {
  long t = (long)blockIdx.x * blockDim.x + threadIdx.x;
  long total = ((long)E + n) << 6;
  if (t >= total) return;
  int e = (int)(t >> 6), c = (int)(t & 63);
  int hd = c >> 5;
  int s, d;
  if (e < E) { s = src[e]; d = dst[e]; } else { s = d = e - E; }
  float ev = gat_logit(as_, ad_, s, d, hd);
  float alpha = __expf(ev - mx[d * 2 + hd]) / (sm[d * 2 + hd] + 1e-16f);
  atomicAdd(&out[((long)d << 6) + c], h[((long)s << 6) + c] * alpha);
}

// ---------------------------------------------------------------------------
// SVI predictor head with fused bias+BN3+ReLU on the input:
//   x = relu(bn3(agg + gcn2_b)); out = sigmoid(relu(x@W1+b1)@W2+b2)
// ---------------------------------------------------------------------------
__global__ __launch_bounds__(256) void head_kernel(
    const float* __restrict__ xin,
    const float* __restrict__ cb, const float* __restrict__ bng,
    const float* __restrict__ bnb, const float* __restrict__ bnm,
    const float* __restrict__ bnv,
    const float* __restrict__ w1, const float* __restrict__ b1,
    const float* __restrict__ w2, const float* __restrict__ b2,
    float* __restrict__ out, int n) {
  __shared__ float sW1[32 * 16], sB1[16], sW2[16], sB2[1];
  __shared__ float sS[32], sT[32];
  int tid = threadIdx.x;
  for (int i = tid; i < 32 * 16; i += 256) sW1[i] = w1[i];
  if (tid < 16) { sB1[tid] = b1[tid]; sW2[tid] = w2[tid]; }
  if (tid < 32) {
    float s = bng[tid] * rsqrtf(bnv[tid] + LN_EPS);
    sS[tid] = s;
    sT[tid] = (cb[tid] - bnm[tid]) * s + bnb[tid];
  }
  if (tid == 0) sB2[0] = b2[0];
  __syncthreads();
  int i = blockIdx.x * 256 + tid;
  if (i >= n) return;
  float xv[32];
#pragma unroll
  for (int j = 0; j < 32; j++)
    xv[j] = fmaxf(xin[i * 32 + j] * sS[j] + sT[j], 0.f);
  float acc2 = sB2[0];
#pragma unroll
  for (int j = 0; j < 16; j++) {
    float s = sB1[j];
#pragma unroll
    for (int k = 0; k < 32; k++) s += xv[k] * sW1[k * 16 + j];
    acc2 += fmaxf(s, 0.f) * sW2[j];
  }
  out[i] = 1.f / (1.f + __expf(-acc2));
}

// ---------------------------------------------------------------------------
// Host orchestration
// ---------------------------------------------------------------------------
static inline int cdiv(long a, long b) { return (int)((a + b - 1) / b); }

extern "C" void kernel_launch(void* const* d_in, const int* in_sizes, int n_in,
                              void* d_out, int out_size, void* d_ws, size_t ws_size,
                              hipStream_t stream) {
  const float* acc     = (const float*)d_in[0];
  const float* ctx     = (const float*)d_in[1];
  const int*   eidx    = (const int*)d_in[2];
  const float* cg_w1   = (const float*)d_in[3];
  const float* cg_b1   = (const float*)d_in[4];
  const float* cg_w2   = (const float*)d_in[5];
  const float* cg_b2   = (const float*)d_in[6];
  const float* cg_aw   = (const float*)d_in[7];
  const float* cg_ab   = (const float*)d_in[8];
  const float* base_imp= (const float*)d_in[9];
  const float* ln_g    = (const float*)d_in[10];
  const float* ln_b    = (const float*)d_in[11];
  const float* fe_w1   = (const float*)d_in[12];
  const float* fe_b1   = (const float*)d_in[13];
  const float* fe_w2   = (const float*)d_in[14];
  const float* fe_b2   = (const float*)d_in[15];
  const float* gcn1_w  = (const float*)d_in[16];
  const float* gcn1_b  = (const float*)d_in[17];
  const float* bn1_g   = (const float*)d_in[18];
  const float* bn1_b   = (const float*)d_in[19];
  const float* bn1_m   = (const float*)d_in[20];
  const float* bn1_v   = (const float*)d_in[21];
  const float* gat_w   = (const float*)d_in[22];
  const float* gat_as  = (const float*)d_in[23];
  const float* gat_ad  = (const float*)d_in[24];
  const float* gat_b   = (const float*)d_in[25];
  const float* bn2_g   = (const float*)d_in[26];
  const float* bn2_b   = (const float*)d_in[27];
  const float* bn2_m   = (const float*)d_in[28];
  const float* bn2_v   = (const float*)d_in[29];
  const float* gcn2_w  = (const float*)d_in[30];
  const float* gcn2_b  = (const float*)d_in[31];
  const float* bn3_g   = (const float*)d_in[32];
  const float* bn3_b   = (const float*)d_in[33];
  const float* bn3_m   = (const float*)d_in[34];
  const float* bn3_v   = (const float*)d_in[35];
  const float* sp_w1   = (const float*)d_in[36];
  const float* sp_b1   = (const float*)d_in[37];
  const float* sp_w2   = (const float*)d_in[38];
  const float* sp_b2   = (const float*)d_in[39];

  const int N = in_sizes[0] / 32;
  const int E = in_sizes[2] / 2;
  const int* src = eidx;
  const int* dst = eidx + E;

  float* ws   = (float*)d_ws;
  float* xln  = ws;                         // [N,32]
  float* bufA = xln  + (size_t)N * 32;      // [N,64]
  float* bufB = bufA + (size_t)N * 64;      // [N,64]
  float* bufH = bufB + (size_t)N * 64;      // [N,64]
  float* deg  = bufH + (size_t)N * 64;      // [N]
  float* as_  = deg  + (size_t)N;           // [N,2]
  float* ad_  = as_  + (size_t)N * 2;       // [N,2]
  float* mxv  = ad_  + (size_t)N * 2;       // [N,2]
  float* smv  = mxv  + (size_t)N * 2;       // [N,2]

  const int TB = 256;
  const int rowTiles = (N + 15) / 16;
  long ETOT = (long)E + N;
  const float* NUL = nullptr;

  // degrees (self-loop included via init=1)
  init_deg_kernel<<<cdiv(N, TB), TB, 0, stream>>>(deg, N);
  count_deg_kernel<<<cdiv(E, TB), TB, 0, stream>>>(dst, deg, E);

  // context gate + LayerNorm
  ctx_gate_ln_kernel<<<cdiv(N, TB), TB, 0, stream>>>(
      acc, ctx, cg_w1, cg_b1, cg_w2, cg_b2, cg_aw, cg_ab, base_imp, ln_g, ln_b, xln, N);

  // feature encoder (WMMA, bias+relu epilogue)
  gemm_wmma_kernel<32, 64, true, true, false><<<cdiv((long)rowTiles * 4, 8), TB, 0, stream>>>(
      xln, fe_w1, fe_b1, bufA, N, NUL, NUL, NUL, NUL, NUL);
  gemm_wmma_kernel<64, 64, true, true, false><<<cdiv((long)rowTiles * 4, 8), TB, 0, stream>>>(
      bufA, fe_w2, fe_b2, bufB, N, NUL, NUL, NUL, NUL, NUL);

  // ---- GCN1: h = x @ W, aggregate into bufA ----
  gemm_wmma_kernel<64, 64, false, false, false><<<cdiv((long)rowTiles * 4, 8), TB, 0, stream>>>(
      bufB, gcn1_w, NUL, bufH, N, NUL, NUL, NUL, NUL, NUL);
  zero_kernel<<<cdiv((long)N * 64, TB), TB, 0, stream>>>(bufA, (long)N * 64);
  gcn_scatter_kernel<<<cdiv(ETOT * 64, TB), TB, 0, stream>>>(bufH, src, dst, deg, bufA, E, N, 6);

  // ---- GAT: input = relu(bn1(bufA + gcn1_b)) fused into A-load ----
  gemm_wmma_kernel<64, 64, false, false, true><<<cdiv((long)rowTiles * 4, 8), TB, 0, stream>>>(
      bufA, gat_w, NUL, bufH, N, gcn1_b, bn1_g, bn1_b, bn1_m, bn1_v);
  gat_alpha_kernel<<<cdiv(N, TB), TB, 0, stream>>>(bufH, gat_as, gat_ad, as_, ad_, N);
  gat_init_kernel<<<cdiv((long)N * 2, TB), TB, 0, stream>>>(mxv, smv, N * 2);
  gat_max_kernel<<<cdiv(ETOT * 2, TB), TB, 0, stream>>>(src, dst, as_, ad_, mxv, E, N);
  gat_sum_kernel<<<cdiv(ETOT * 2, TB), TB, 0, stream>>>(src, dst, as_, ad_, mxv, smv, E, N);
  zero_kernel<<<cdiv((long)N * 64, TB), TB, 0, stream>>>(bufA, (long)N * 64);
  gat_scatter_kernel<<<cdiv(ETOT * 64, TB), TB, 0, stream>>>(
      bufH, src, dst, as_, ad_, mxv, smv, bufA, E, N);

  // ---- GCN2 (64->32): input = relu(bn2(bufA + gat_b)) fused into A-load ----
  gemm_wmma_kernel<64, 32, false, false, true><<<cdiv((long)rowTiles * 2, 8), TB, 0, stream>>>(
      bufA, gcn2_w, NUL, bufH, N, gat_b, bn2_g, bn2_b, bn2_m, bn2_v);
  zero_kernel<<<cdiv((long)N * 32, TB), TB, 0, stream>>>(bufB, (long)N * 32);
  gcn_scatter_kernel<<<cdiv(ETOT * 32, TB), TB, 0, stream>>>(bufH, src, dst, deg, bufB, E, N, 5);

  // ---- head: input = relu(bn3(bufB + gcn2_b)) fused ----
  head_kernel<<<cdiv(N, TB), TB, 0, stream>>>(bufB, gcn2_b, bn3_g, bn3_b, bn3_m, bn3_v,
                                              sp_w1, sp_b1, sp_w2, sp_b2,
                                              (float*)d_out, N);
}